// EdgeConsistencyLoss_7017976562410
// MI455X (gfx1250) — compile-verified
//
#include <hip/hip_runtime.h>
#include <hip/hip_bf16.h>
#include <stdint.h>

typedef __attribute__((ext_vector_type(2))) float v2f;
typedef __attribute__((ext_vector_type(8))) float v8f;

#define IMG_H 512
#define IMG_W 512
#define PLANES 128          // B*C = 16*8
#define TILE_R 16
#define TILE_C 32
#define LDS_R 18            // TILE_R + 2 halo rows
#define LDS_C 40            // TILE_C + halo, padded to 16B-aligned chunks
#define LDS_ELEMS (LDS_R * LDS_C)    // 720 floats
#define CHUNKS (LDS_R * (LDS_C / 4)) // 180 float4 chunks per tensor tile
#define WAVES_PER_BLOCK 8
#define INV_N (1.0f / 33554432.0f)   // 1 / (16*8*512*512)

__global__ void zero_out_kernel(float* out) { out[0] = 0.0f; }

// xor-swizzle add within wave32 (group-of-32 mode: offset = xor<<10 | and 0x1f)
template <int XOR>
__device__ inline float swz_xor_add(float v) {
    int j = __builtin_amdgcn_ds_swizzle(__float_as_int(v), (XOR << 10) | 0x1f);
    return v + __int_as_float(j);
}

__global__ __launch_bounds__(256) void sobel_edge_loss_kernel(
    const float* __restrict__ src, const float* __restrict__ edt,
    float* __restrict__ out)
{
    __shared__ __align__(16) float smem[WAVES_PER_BLOCK][2][LDS_ELEMS];

    const int lane = threadIdx.x & 31;
    const int wave = threadIdx.x >> 5;
    const int w    = blockIdx.x * WAVES_PER_BLOCK + wave;   // global wave id
    const int plane = w >> 9;            // 512 tiles per plane
    const int rem   = w & 511;
    const int row0  = (rem >> 4) * TILE_R;   // 32 row-tiles
    const int col0  = (rem & 15) * TILE_C;   // 16 col-tiles
    const size_t pbase = (size_t)plane * (IMG_H * IMG_W);

    // ---- async bulk copy: global -> LDS (gfx1250 ASYNCcnt path) ----
    #pragma unroll
    for (int t = 0; t < 2; ++t) {
        const float* gp = (t == 0 ? src : edt) + pbase;
        const uint32_t lds_base = (uint32_t)(uintptr_t)&smem[wave][t][0];
        #pragma unroll
        for (int i = 0; i < 6; ++i) {
            int chunk = i * 32 + lane;
            if (chunk < CHUNKS) {
                int r  = chunk / (LDS_C / 4);
                int kc = chunk % (LDS_C / 4);
                int gr = row0 - 1 + r;
                gr = gr < 0 ? 0 : (gr > IMG_H - 1 ? IMG_H - 1 : gr);
                int gc = col0 - 4 + kc * 4;
                gc = gc < 0 ? 0 : (gc > IMG_W - 4 ? IMG_W - 4 : gc);
                const float* ga = gp + (size_t)gr * IMG_W + gc;
                uint32_t la = lds_base + (uint32_t)(r * LDS_C + kc * 4) * 4u;
                asm volatile("global_load_async_to_lds_b128 %0, %1, off"
                             :: "v"(la), "v"(ga) : "memory");
            }
        }
    }
    asm volatile("s_wait_asynccnt 0x0" ::: "memory");

    // ---- SAME zero-padding fixup (border tiles only; uniform branch) ----
    if (row0 == 0 || row0 + TILE_R == IMG_H || col0 == 0 || col0 + TILE_C == IMG_W) {
        for (int idx = lane; idx < LDS_ELEMS; idx += 32) {
            int r = idx / LDS_C, c = idx % LDS_C;
            int gr = row0 - 1 + r, gc = col0 - 4 + c;
            if (gr < 0 || gr >= IMG_H || gc < 0 || gc >= IMG_W) {
                smem[wave][0][idx] = 0.0f;
                smem[wave][1][idx] = 0.0f;
            }
        }
    }

    // ---- separable Sobel via row prefilters (hx = diff, hy = smooth) ----
    // gx(r) = hx(r-1) + 2*hx(r) + hx(r+1);  gy(r) = hy(r+1) - hy(r-1)
    const float* lp[2] = { &smem[wave][0][0], &smem[wave][1][0] };
    const int lc = 4 + lane;             // this lane's LDS column

    float hx[2][2], hy[2][2];            // rolling prefilters, rows r-1, r
    #pragma unroll
    for (int t = 0; t < 2; ++t)
        #pragma unroll
        for (int rr = 0; rr < 2; ++rr) {
            float a = lp[t][rr * LDS_C + lc - 1];
            float b = lp[t][rr * LDS_C + lc];
            float c = lp[t][rr * LDS_C + lc + 1];
            hx[t][rr] = c - a;
            hy[t][rr] = __builtin_fmaf(2.0f, b, a + c);
        }

    float acc = 0.0f;
    #pragma unroll
    for (int r = 0; r < TILE_R; ++r) {
        float e[2];
        #pragma unroll
        for (int t = 0; t < 2; ++t) {
            float a = lp[t][(r + 2) * LDS_C + lc - 1];
            float b = lp[t][(r + 2) * LDS_C + lc];
            float c = lp[t][(r + 2) * LDS_C + lc + 1];
            float hxn = c - a;
            float hyn = __builtin_fmaf(2.0f, b, a + c);
            float gx  = __builtin_fmaf(2.0f, hx[t][1], hx[t][0] + hxn);
            float gy  = hyn - hy[t][0];
            e[t] = __builtin_amdgcn_sqrtf(
                       __builtin_fmaf(gx, gx, __builtin_fmaf(gy, gy, 1e-8f)));
            hx[t][0] = hx[t][1]; hx[t][1] = hxn;   // renamed away by unroll
            hy[t][0] = hy[t][1]; hy[t][1] = hyn;
        }
        float d = e[1] - e[0];
        acc = __builtin_fmaf(d, d, acc);
    }

    // ---- wave reduction via V_WMMA_F32_16X16X4_F32 ----
    // A = ones(16x4), B[0][j]=acc_j, B[2][j]=acc_{j+16}
    // => every row of D holds colsum_j = acc_j + acc_{j+16}.
    v2f a; a[0] = 1.0f; a[1] = 1.0f;
    v2f b; b[0] = acc;  b[1] = 0.0f;
    v8f c = {};
    c = __builtin_amdgcn_wmma_f32_16x16x4_f32(
            /*neg_a=*/false, a, /*neg_b=*/false, b,
            /*c_mod=*/(short)0, c, /*reuse_a=*/false, /*reuse_b=*/false);

    float s = c[0];                 // lane j and j+16 both hold colsum_j
    s = swz_xor_add<1>(s);
    s = swz_xor_add<2>(s);
    s = swz_xor_add<4>(s);
    s = swz_xor_add<8>(s);          // all lanes now hold the full wave sum

    if (lane == 0)
        atomicAdd(out, s * INV_N);
}

extern "C" void kernel_launch(void* const* d_in, const int* in_sizes, int n_in,
                              void* d_out, int out_size, void* d_ws, size_t ws_size,
                              hipStream_t stream) {
    (void)in_sizes; (void)n_in; (void)d_ws; (void)ws_size; (void)out_size;
    const float* src = (const float*)d_in[0];   // source_latent [16,8,512,512]
    const float* edt = (const float*)d_in[1];   // edited_latent [16,8,512,512]
    // d_in[2]/d_in[3] are the fixed Sobel taps; they are baked into the kernel.
    float* out = (float*)d_out;

    zero_out_kernel<<<1, 1, 0, stream>>>(out);

    const int total_waves = PLANES * (IMG_H / TILE_R) * (IMG_W / TILE_C); // 65536
    const int blocks = total_waves / WAVES_PER_BLOCK;                     // 8192
    sobel_edge_loss_kernel<<<blocks, 256, 0, stream>>>(src, edt, out);
}